// GCNForwardModel_73589969649974
// MI455X (gfx1250) — compile-verified
//
#include <hip/hip_runtime.h>

#define NNODES  50000
#define NEDGES  800000
#define DIM     128
#define NLAYERS 3

typedef __attribute__((ext_vector_type(2))) float v2f;
typedef __attribute__((ext_vector_type(8))) float v8f;

// Explicit CDNA5 device-scope fp32 atomic add (no return -> STOREcnt path).
// S_ENDPGM performs an implicit wait-idle, so completion is guaranteed at
// kernel end without a manual s_wait_storecnt.
__device__ __forceinline__ void atomic_add_f32_dev(float* p, float v) {
    asm volatile("global_atomic_add_f32 %0, %1, off scope:SCOPE_DEV"
                 :: "v"(p), "v"(v) : "memory");
}

// ---------------------------------------------------------------------------
// Zero the aggregation buffer (graph-capture-safe, no memset).
// ---------------------------------------------------------------------------
__global__ void gcn_zero_kernel(float4* __restrict__ p, int n4) {
    int i = blockIdx.x * blockDim.x + threadIdx.x;
    if (i < n4) p[i] = make_float4(0.f, 0.f, 0.f, 0.f);
}

// ---------------------------------------------------------------------------
// Edge phase: agg[dst] += w * h[src].  One wave32 per edge, float4 per lane
// (32 lanes * 4 floats = 128 = DIM) -> fully coalesced 512B gather per edge.
// h (25.6MB) is L2-resident on MI455X (192MB L2), so this is L2-bound.
// ---------------------------------------------------------------------------
__global__ __launch_bounds__(256) void gcn_scatter_kernel(
    const float* __restrict__ h,
    const int*   __restrict__ edge_index,   // [2, E]
    const float* __restrict__ edge_weight,  // [E]
    float*       __restrict__ agg,          // [N, DIM]
    int nedges) {
    int tid  = blockIdx.x * blockDim.x + threadIdx.x;
    int e    = tid >> 5;
    int lane = tid & 31;
    if (e >= nedges) return;

    int   src = edge_index[e];
    int   dst = edge_index[nedges + e];
    float w   = edge_weight[e];

    const float4* hv = reinterpret_cast<const float4*>(h + (size_t)src * DIM);
    float4 v = hv[lane];

    float* a = agg + (size_t)dst * DIM + (size_t)lane * 4;
    atomic_add_f32_dev(a + 0, v.x * w);
    atomic_add_f32_dev(a + 1, v.y * w);
    atomic_add_f32_dev(a + 2, v.z * w);
    atomic_add_f32_dev(a + 3, v.w * w);
}

// ---------------------------------------------------------------------------
// Node phase: out = agg @ Wrel + h @ Wroot + b  (optional ReLU), fp32 WMMA.
// Block = 256 threads = 8 wave32.  Block owns a 16-row node tile; wave `w`
// owns output columns [16w, 16w+16).
//
// The 16x128 agg and h tiles are staged in LDS once per block (16KB of the
// 320KB WGP LDS) -> removes the 8x redundant per-wave global A loads; inner
// loop A reads become ds_load_b64.  Two independent accumulators (rel/root)
// break the single WMMA C->D dependency chain.
//
// ISA VGPR layouts (CDNA5 ISA):
//   A 16x4 : lane l -> row M=l%16, v0 = K=2*(l/16), v1 = K=2*(l/16)+1
//   B 4x16 : lane l -> col N=l%16, v0 = K=2*(l/16), v1 = K=2*(l/16)+1
//   C/D    : lane l, VGPR j -> element [M = j + 8*(l/16)][N = l%16]
// N=50000 is a multiple of 16 -> EXEC all-ones through the WMMAs.
// ---------------------------------------------------------------------------
__global__ __launch_bounds__(256) void gcn_gemm_wmma_kernel(
    const float* __restrict__ agg,    // [N, DIM]
    const float* __restrict__ h,      // [N, DIM]
    const float* __restrict__ Wrel,   // [DIM, DIM] row-major
    const float* __restrict__ Wroot,  // [DIM, DIM] row-major
    const float* __restrict__ bias,   // [DIM]
    float*       __restrict__ out,    // [N, DIM]
    int nrows, int do_relu) {
    __shared__ float4 sA4[16 * DIM / 4];   // agg tile, 16 rows x 128 cols
    __shared__ float4 sH4[16 * DIM / 4];   // h   tile

    const int t    = threadIdx.x;
    const int lane = t & 31;
    const int wv   = t >> 5;              // 0..7 -> column tile
    const int r    = lane & 15;
    const int half = lane >> 4;           // 0 or 1

    const int row0 = blockIdx.x * 16;
    const int col0 = wv * 16;

    // ---- cooperative stage of A tiles into LDS (2 float4 per thread each) --
    {
        const float4* aggv = reinterpret_cast<const float4*>(agg) + (size_t)row0 * (DIM / 4);
        const float4* hv   = reinterpret_cast<const float4*>(h)   + (size_t)row0 * (DIM / 4);
#pragma unroll
        for (int i = t; i < 16 * DIM / 4; i += 256) {
            sA4[i] = aggv[i];
            sH4[i] = hv[i];
        }
    }
    __syncthreads();

    const float* sA = reinterpret_cast<const float*>(sA4);
    const float* sH = reinterpret_cast<const float*>(sH4);

    v8f acc_rel  = {};
    v8f acc_root = {};
#pragma unroll
    for (int k = 0; k < DIM; k += 4) {
        const int ka = k + 2 * half;      // this lane's K pair

        v2f a_rel  = *reinterpret_cast<const v2f*>(&sA[r * DIM + ka]);  // ds_load_b64
        v2f b_rel  = { Wrel[(size_t)ka * DIM + col0 + r],
                       Wrel[(size_t)(ka + 1) * DIM + col0 + r] };
        acc_rel = __builtin_amdgcn_wmma_f32_16x16x4_f32(
                      false, a_rel, false, b_rel, (short)0, acc_rel, false, false);

        v2f a_root = *reinterpret_cast<const v2f*>(&sH[r * DIM + ka]);  // ds_load_b64
        v2f b_root = { Wroot[(size_t)ka * DIM + col0 + r],
                       Wroot[(size_t)(ka + 1) * DIM + col0 + r] };
        acc_root = __builtin_amdgcn_wmma_f32_16x16x4_f32(
                      false, a_root, false, b_root, (short)0, acc_root, false, false);
    }

    const float bv = bias[col0 + r];
#pragma unroll
    for (int j = 0; j < 8; ++j) {
        const int m    = j + 8 * half;
        const int orow = row0 + m;
        if (orow < nrows) {
            float v = acc_rel[j] + acc_root[j] + bv;
            if (do_relu) v = fmaxf(v, 0.f);
            out[(size_t)orow * DIM + col0 + r] = v;
        }
    }
}

// ---------------------------------------------------------------------------
// Driver: 3 GraphConv layers.  Workspace: h1 | h2 | agg  (3 * N * DIM floats).
// ---------------------------------------------------------------------------
extern "C" void kernel_launch(void* const* d_in, const int* in_sizes, int n_in,
                              void* d_out, int out_size, void* d_ws, size_t ws_size,
                              hipStream_t stream) {
    const float* x      = (const float*)d_in[0];   // [N, DIM]
    const int*   eidx   = (const int*)  d_in[1];   // [2, E]
    const float* ew     = (const float*)d_in[2];   // [E]
    const float* Wrel   = (const float*)d_in[3];   // [L, DIM, DIM]
    const float* brel   = (const float*)d_in[4];   // [L, DIM]
    const float* Wroot  = (const float*)d_in[5];   // [L, DIM, DIM]
    float*       out    = (float*)d_out;

    const int N = NNODES;
    const int E = NEDGES;

    float* h1  = (float*)d_ws;
    float* h2  = h1 + (size_t)N * DIM;
    float* agg = h2 + (size_t)N * DIM;

    const int n4          = N * DIM / 4;
    const int zero_blocks = (n4 + 255) / 256;
    const int scat_blocks = (E * 32 + 255) / 256;
    const int gemm_blocks = (N + 15) / 16;

    for (int l = 0; l < NLAYERS; ++l) {
        const float* hin  = (l == 0) ? x  : (l == 1) ? h1 : h2;
        float*       hout = (l == 0) ? h1 : (l == 1) ? h2 : out;

        gcn_zero_kernel<<<zero_blocks, 256, 0, stream>>>(
            (float4*)agg, n4);

        gcn_scatter_kernel<<<scat_blocks, 256, 0, stream>>>(
            hin, eidx, ew, agg, E);

        gcn_gemm_wmma_kernel<<<gemm_blocks, 256, 0, stream>>>(
            agg, hin,
            Wrel  + (size_t)l * DIM * DIM,
            Wroot + (size_t)l * DIM * DIM,
            brel  + (size_t)l * DIM,
            hout, N, (l < NLAYERS - 1) ? 1 : 0);
    }
}